// MultVariateKLD_85968065397106
// MI455X (gfx1250) — compile-verified
//
#include <hip/hip_runtime.h>

// ---------------------------------------------------------------------------
// MultVariateKLD (diagonal-Gaussian KL, mean over batch) for MI455X / gfx1250
//
// Memory-bound streaming reduction: 335 MB in, 4 B out  ->  ~14.4 us floor
// at 23.3 TB/s.  Strategy: one fused pass with NT b128 loads, per-thread f32
// partials, wave-level reduction via V_WMMA_F32_16X16X4_F32 (B = ones),
// LDS across the 8 wave32s of a block, then a deterministic f64 final pass.
// ---------------------------------------------------------------------------

typedef __attribute__((ext_vector_type(4))) float fvec4;
typedef __attribute__((ext_vector_type(2))) float v2f;
typedef __attribute__((ext_vector_type(8))) float v8f;

#define KL_BLOCKS  4096
#define KL_THREADS 256
#define KL_WAVES   (KL_THREADS / 32)

// Cross-lane (wave32) sum using the CDNA5 matrix unit.
// A is 16x4 f32: VGPR0 holds (M=lane,K=0) for lanes 0-15 and (M=lane-16,K=2)
// for lanes 16-31; VGPR1 is K=1/K=3 which we zero.  B = 4x16 of ones, so
// D[m][n] = x_m + x_{m+16}.  Summing the 8 D VGPRs gives half-wave sums in
// lanes 0-15 / 16-31; one shfl_xor(16) completes the full wave sum.
__device__ __forceinline__ float wave_sum_wmma(float x) {
  v2f a;
  a.x = x;
  a.y = 0.0f;
  v2f b;
  b.x = 1.0f;
  b.y = 1.0f;
  v8f c = {};
  c = __builtin_amdgcn_wmma_f32_16x16x4_f32(
      /*neg_a=*/false, a, /*neg_b=*/false, b,
      /*c_mod=*/(short)0, c, /*reuse_a=*/false, /*reuse_b=*/false);
  float s = c[0] + c[1] + c[2] + c[3] + c[4] + c[5] + c[6] + c[7];
  s += __shfl_xor(s, 16, 32);
  return s;  // every lane holds the wave total
}

__global__ void __launch_bounds__(KL_THREADS)
kl_partial_kernel(const fvec4* __restrict__ mu1, const fvec4* __restrict__ mu2,
                  const fvec4* __restrict__ sg1, const fvec4* __restrict__ sg2,
                  const fvec4* __restrict__ msk, float* __restrict__ partials,
                  int nvec) {
  const int stride = gridDim.x * blockDim.x;
  float acc_log2 = 0.0f;  // sum of log2(s2) - log2(s1)  (scaled by ln2 later)
  float acc_lin  = 0.0f;  // sum of (s1 + d*d)/s2 - 1

  for (int i = blockIdx.x * blockDim.x + threadIdx.x; i < nvec; i += stride) {
    // Use-once stream: non-temporal b128 loads, keep the 192MB L2 clean.
    fvec4 m1 = __builtin_nontemporal_load(&mu1[i]);
    fvec4 m2 = __builtin_nontemporal_load(&mu2[i]);
    fvec4 s1 = __builtin_nontemporal_load(&sg1[i]);
    fvec4 s2 = __builtin_nontemporal_load(&sg2[i]);
    fvec4 mk = __builtin_nontemporal_load(&msk[i]);
#pragma unroll
    for (int c = 0; c < 4; ++c) {
      float a = m1[c];
      a = (a != a) ? 0.0f : a;  // jnp.nan_to_num(mu1) (inputs are finite)
      float d = mk[c] * (m2[c] - a);
      acc_log2 += __log2f(s2[c]) - __log2f(s1[c]);  // v_log_f32 path
      acc_lin  += (s1[c] + d * d) / s2[c] - 1.0f;
    }
  }

  // Per-thread partial (convert log2 units -> ln once per thread).
  float p = acc_lin + 0.69314718055994531f * acc_log2;

  // Wave32 reduction on the matrix unit (EXEC is all-1s here: no divergence).
  float wsum = wave_sum_wmma(p);

  __shared__ float lds[KL_WAVES];
  const int wave = threadIdx.x >> 5;
  const int lane = threadIdx.x & 31;
  if (lane == 0) lds[wave] = wsum;
  __syncthreads();
  if (threadIdx.x == 0) {
    float t = 0.0f;
#pragma unroll
    for (int w = 0; w < KL_WAVES; ++w) t += lds[w];
    partials[blockIdx.x] = t;
  }
}

// Deterministic final pass: one block accumulates the 4096 block partials in
// fp64 (matches the fp64 reference reduction) and applies 0.5/B.
__global__ void __launch_bounds__(KL_THREADS)
kl_final_kernel(const float* __restrict__ partials, int n,
                float* __restrict__ out, double scale) {
  double acc = 0.0;
  for (int i = threadIdx.x; i < n; i += blockDim.x)
    acc += (double)partials[i];

#pragma unroll
  for (int off = 16; off > 0; off >>= 1)
    acc += __shfl_down(acc, off, 32);

  __shared__ double lds[KL_WAVES];
  const int wave = threadIdx.x >> 5;
  const int lane = threadIdx.x & 31;
  if (lane == 0) lds[wave] = acc;
  __syncthreads();
  if (threadIdx.x == 0) {
    double t = 0.0;
#pragma unroll
    for (int w = 0; w < KL_WAVES; ++w) t += lds[w];
    out[0] = (float)(t * scale);
  }
}

extern "C" void kernel_launch(void* const* d_in, const int* in_sizes, int n_in,
                              void* d_out, int out_size, void* d_ws, size_t ws_size,
                              hipStream_t stream) {
  const fvec4* mu1 = (const fvec4*)d_in[0];
  const fvec4* mu2 = (const fvec4*)d_in[1];
  const fvec4* sg1 = (const fvec4*)d_in[2];
  const fvec4* sg2 = (const fvec4*)d_in[3];
  const fvec4* msk = (const fvec4*)d_in[4];

  const int total = in_sizes[0];     // B*N = 65536*256 = 16,777,216
  const int nvec  = total / 4;       // float4 granules
  const int N     = 256;             // feature dim (reference constant)
  const int Bsz   = total / N;

  float* partials = (float*)d_ws;    // KL_BLOCKS floats (16 KB) << ws_size
  float* out      = (float*)d_out;

  kl_partial_kernel<<<KL_BLOCKS, KL_THREADS, 0, stream>>>(
      mu1, mu2, sg1, sg2, msk, partials, nvec);

  const double scale = 0.5 / (double)Bsz;
  kl_final_kernel<<<1, KL_THREADS, 0, stream>>>(partials, KL_BLOCKS, out, scale);
}